// NCRF_34248069218452
// MI455X (gfx1250) — compile-verified
//
#include <hip/hip_runtime.h>

// Mean-field CRF on MI455X (gfx1250).
// Mathematical reduction: einsum('bnhw,bcdhw->bchw') factorizes into
// (sum_n q)*(sum_d pair); softmax sums to 1, so the update is constant across
// all 10 iterations:   out = unary - sum_{d,kx,ky} clip(pair, 0.001, 100).
// Pure bandwidth-bound streaming reduction (~310 MB -> ~13 us at 23.3 TB/s).
// The 72-plane reduction is fed through v_wmma_f32_16x16x4_f32 with B = ones
// (D[m,n] += sum_k A[m,k]), which is exact fp32 accumulation and free on the
// matrix pipe while the kernel streams from HBM.

typedef __attribute__((ext_vector_type(2))) float v2f;
typedef __attribute__((ext_vector_type(8))) float v8f;

#define HW      65536   // 256*256
#define NPLANES 72      // C * K * K = 8 * 3 * 3
#define NCHUNK  18      // 72 / 4  (K=4 per WMMA)

__device__ __forceinline__ float clip_cdna5(float x) {
    return fminf(fmaxf(x, 0.001f), 100.0f);
}

// Select element e (0..7) of a v8f without scratch: cndmask chain.
__device__ __forceinline__ float pick8(v8f v, int e) {
    float s = v[0];
#pragma unroll
    for (int r = 1; r < 8; ++r) s = (e == r) ? v[r] : s;
    return s;
}

__global__ __launch_bounds__(256) void NCRF_crf_collapse_kernel(
    const float* __restrict__ unary,   // (B,C,H,W)      = 16 * 65536
    const float* __restrict__ pair,    // (B,C,C,3,3,H,W)= 16 * 72 * 65536
    float* __restrict__ out)           // (B,C,H,W)
{
    const int tid  = blockIdx.x * blockDim.x + threadIdx.x;
    const int wave = tid >> 5;
    const int lane = threadIdx.x & 31;
    const int half = lane >> 4;   // 0: K={0,1}, 1: K={2,3}
    const int m    = lane & 15;   // A-matrix row (spatial position in tile)

    // Each wave produces 32 consecutive outputs: two 16-wide WMMA tiles.
    const int g    = wave << 5;           // first output index of this wave
    const int slab = g >> 16;             // (b*C + c), HW = 65536 divides g-range
    const int hw   = g & (HW - 1);

    // Per-lane base pointer into the 72 planes of this slab.
    const float* p0 = pair + (size_t)slab * NPLANES * HW
                           + (size_t)(2 * half) * HW + hw + m;
    const float* p1 = p0 + 16;            // second tile, adjacent 16 pixels

    v8f acc0 = {};
    v8f acc1 = {};
    v2f ones; ones.x = 1.0f; ones.y = 1.0f;   // B matrix = all ones (4x16)

#pragma unroll
    for (int t = 0; t < NCHUNK; ++t) {
        // A operand: lane(l<16) -> {K=4t, K=4t+1}; lane(l>=16) -> {K=4t+2, 4t+3}
        float a0x = clip_cdna5(p0[0]);
        float a0y = clip_cdna5(p0[HW]);
        float a1x = clip_cdna5(p1[0]);
        float a1y = clip_cdna5(p1[HW]);
        v2f a0; a0.x = a0x; a0.y = a0y;
        v2f a1; a1.x = a1x; a1.y = a1y;
        // D = A x ones + D  ==>  per-row sum of the 4 planes, fp32 exact.
        acc0 = __builtin_amdgcn_wmma_f32_16x16x4_f32(
            false, a0, false, ones, (short)0, acc0, false, false);
        acc1 = __builtin_amdgcn_wmma_f32_16x16x4_f32(
            false, a1, false, ones, (short)0, acc1, false, false);
        p0 += 4 * HW;
        p1 += 4 * HW;
    }

    // D layout: VGPR r | lanes 0-15: M=r | lanes 16-31: M=8+r  (any N column).
    // Assign every lane exactly one of the 32 outputs -> coalesced 128B store.
    const int grp = lane >> 3;        // 0..3
    const int k   = lane & 7;
    const int mm  = ((grp >> 1) << 3) + k;          // grp 0,1 -> m=k; 2,3 -> m=8+k
    const int tofs = (grp & 1) ? 16 : 0;            // odd groups take tile 1
    const float s = (grp & 1) ? pick8(acc1, k) : pick8(acc0, k);

    const int idx = g + tofs + mm;
    out[idx] = unary[idx] - s;
}

extern "C" void kernel_launch(void* const* d_in, const int* in_sizes, int n_in,
                              void* d_out, int out_size, void* d_ws, size_t ws_size,
                              hipStream_t stream) {
    (void)in_sizes; (void)n_in; (void)d_ws; (void)ws_size;
    const float* unary = (const float*)d_in[0];   // (2,8,256,256) fp32
    const float* pair  = (const float*)d_in[1];   // (2,8,8,3,3,256,256) fp32
    float* out = (float*)d_out;                   // (2,8,256,256) fp32

    // One thread per output element: 16 slabs * 65536 = 1,048,576 threads.
    // 32 outputs per wave (two 16-wide WMMA tiles), 256 threads/block.
    const int total  = out_size;                  // 1,048,576
    const int block  = 256;
    const int blocks = total / block;             // 4096
    hipLaunchKernelGGL(NCRF_crf_collapse_kernel, dim3(blocks), dim3(block), 0,
                       stream, unary, pair, out);
}